// GINModel_19877108646248
// MI455X (gfx1250) — compile-verified
//
#include <hip/hip_runtime.h>
#include <hip/hip_bf16.h>

typedef __attribute__((ext_vector_type(2))) float v2f;
typedef __attribute__((ext_vector_type(8))) float v8f;

#define HID 128
#define LDP 132   // LDS row pitch (floats), avoids bank conflicts (132 % 64 = 4)

// ---------------------------------------------------------------------------
// Wave-level 16x16 output tile GEMM over K=128 using V_WMMA_F32_16X16X4_F32.
// A tile comes from LDS (16 rows x 128 cols, pitch LDP).
// W is [128 x 128] row-major (k-major, as in z @ W).
// Each wave owns output columns [nBase, nBase+16).
//
// Fragment layouts per CDNA5 ISA 7.12.2 (32-bit, wave32):
//   A 16x4 : lanes 0-15 -> {K0,K1}, lanes 16-31 -> {K2,K3}; row M = lane%16
//   B 4x16 : VGPR0/1 mirror A (row K = k0 + 2*(lane/16) + {0,1}); col N = lane%16
//   C 16x16: VGPR r, element (M = r + 8*(lane/16), N = nBase + lane%16)
// ---------------------------------------------------------------------------
template <bool RELU, bool TO_LDS>
__device__ __forceinline__ void gemm16_tile(const float* __restrict__ lA,
                                            const float* __restrict__ W,
                                            const float* __restrict__ bias,
                                            float* __restrict__ dstLds,
                                            float* __restrict__ dstGlb,
                                            int rowBase, int rows,
                                            int lane, int nBase) {
    const int hi = lane >> 4;        // 0 or 1
    const int lo = lane & 15;
    const int n  = nBase + lo;
    v8f acc = {};
#pragma unroll
    for (int k0 = 0; k0 < HID; k0 += 4) {
        const int kk = k0 + 2 * hi;
        // A fragment: 2 consecutive K values of row `lo` from LDS
        const float2 av = *(const float2*)(lA + lo * LDP + kk);
        v2f a; a.x = av.x; a.y = av.y;
        // B fragment: rows kk, kk+1 of W at column n
        v2f b;
        b.x = W[(kk + 0) * HID + n];
        b.y = W[(kk + 1) * HID + n];
        acc = __builtin_amdgcn_wmma_f32_16x16x4_f32(
            /*neg_a=*/false, a, /*neg_b=*/false, b,
            /*c_mod=*/(short)0, acc, /*reuse_a=*/false, /*reuse_b=*/false);
    }
    const float bv = bias[n];
#pragma unroll
    for (int r = 0; r < 8; ++r) {
        const int m = r + 8 * hi;
        float v = acc[r] + bv;
        if (RELU) v = fmaxf(v, 0.0f);
        if (TO_LDS) {
            dstLds[m * LDP + n] = v;
        } else {
            const int row = rowBase + m;
            if (row < rows) dstGlb[(size_t)row * HID + n] = v;
        }
    }
}

// ---------------------------------------------------------------------------
// Fused GIN MLP for hidden layers: out = relu((h+agg) @ W1 + b1) @ W2 + b2
// One block = 16 rows x 128 cols, 8 waves.
// ---------------------------------------------------------------------------
__global__ __launch_bounds__(256) void mlp_kernel(
    const float* __restrict__ Hin, const float* __restrict__ Agg,
    const float* __restrict__ W1, const float* __restrict__ b1,
    const float* __restrict__ W2, const float* __restrict__ b2,
    float* __restrict__ Out, int rows) {
    __shared__ float zt[16 * LDP];
    __shared__ float tt[16 * LDP];
    const int rowBase = blockIdx.x * 16;
    const int tid = threadIdx.x;

    // Stage 1: z = h + agg into LDS (zero-fill rows past end)
    for (int i = tid; i < 16 * HID; i += 256) {
        const int m = i >> 7, c = i & (HID - 1);
        const int row = rowBase + m;
        float v = 0.0f;
        if (row < rows) {
            const size_t off = (size_t)row * HID + c;
            v = Hin[off] + Agg[off];
        }
        zt[m * LDP + c] = v;
    }
    __syncthreads();

    const int wave = tid >> 5, lane = tid & 31;
    // Stage 2: t = relu(z @ W1 + b1) -> LDS
    gemm16_tile<true, true>(zt, W1, b1, tt, nullptr, rowBase, rows, lane, wave * 16);
    __syncthreads();
    // Stage 3: out = t @ W2 + b2 -> global
    gemm16_tile<false, false>(tt, W2, b2, nullptr, Out, rowBase, rows, lane, wave * 16);
}

// ---------------------------------------------------------------------------
// Layer 0: din = 1. t = relu(z0 * W1[1x128] + b1) (outer product, elementwise),
// then out = t @ W2 + b2 via WMMA.
// ---------------------------------------------------------------------------
__global__ __launch_bounds__(256) void layer0_kernel(
    const float* __restrict__ x, const float* __restrict__ agg0,
    const float* __restrict__ W1, const float* __restrict__ b1,
    const float* __restrict__ W2, const float* __restrict__ b2,
    float* __restrict__ Out, int rows) {
    __shared__ float tt[16 * LDP];
    __shared__ float z0[16];
    const int rowBase = blockIdx.x * 16;
    const int tid = threadIdx.x;

    if (tid < 16) {
        const int row = rowBase + tid;
        z0[tid] = (row < rows) ? (x[row] + agg0[row]) : 0.0f;
    }
    __syncthreads();
    for (int i = tid; i < 16 * HID; i += 256) {
        const int m = i >> 7, n = i & (HID - 1);
        tt[m * LDP + n] = fmaxf(z0[m] * W1[n] + b1[n], 0.0f);
    }
    __syncthreads();
    gemm16_tile<false, false>(tt, W2, b2, nullptr, Out, rowBase, rows,
                              tid & 31, (tid >> 5) * 16);
}

// ---------------------------------------------------------------------------
// Single linear (final FC): out = A @ W + b   (A: [rows,128], no relu)
// ---------------------------------------------------------------------------
__global__ __launch_bounds__(256) void linear_kernel(
    const float* __restrict__ A, const float* __restrict__ W,
    const float* __restrict__ bias, float* __restrict__ Out, int rows) {
    __shared__ float at[16 * LDP];
    const int rowBase = blockIdx.x * 16;
    const int tid = threadIdx.x;
    for (int i = tid; i < 16 * HID; i += 256) {
        const int m = i >> 7, c = i & (HID - 1);
        const int row = rowBase + m;
        at[m * LDP + c] = (row < rows) ? A[(size_t)row * HID + c] : 0.0f;
    }
    __syncthreads();
    gemm16_tile<false, false>(at, W, bias, nullptr, Out, rowBase, rows,
                              tid & 31, (tid >> 5) * 16);
}

// ---------------------------------------------------------------------------
// Sparse kernels
// ---------------------------------------------------------------------------
__global__ void zero_kernel(float* p, long long n) {
    const long long i = (long long)blockIdx.x * blockDim.x + threadIdx.x;
    if (i < n) p[i] = 0.0f;
}

// Layer 0 scatter (feature dim = 1): agg0[dst] += x[src]
__global__ void scatter0_kernel(const float* __restrict__ x,
                                const int* __restrict__ ei,
                                float* __restrict__ agg0, int nE) {
    const int e = blockIdx.x * blockDim.x + threadIdx.x;
    if (e < nE) {
        const int s = ei[e];
        const int d = ei[nE + e];
        atomicAdd(&agg0[d], x[s]);
    }
}

// Hidden scatter: one wave per edge, 32 lanes x float4 covers the 512B row.
__global__ void scatterH_kernel(const float* __restrict__ h,
                                const int* __restrict__ ei,
                                float* __restrict__ agg, int nE) {
    const long long t = (long long)blockIdx.x * blockDim.x + threadIdx.x;
    const int e = (int)(t >> 5);
    const int c = (int)(t & 31);
    if (e < nE) {
        const int s = ei[e];
        const int d = ei[nE + e];
        const float4 v = *(const float4*)(h + (size_t)s * HID + c * 4);
        float* dp = agg + (size_t)d * HID + c * 4;
        atomicAdd(dp + 0, v.x);
        atomicAdd(dp + 1, v.y);
        atomicAdd(dp + 2, v.z);
        atomicAdd(dp + 3, v.w);
    }
}

// Global add pool: pooled[batch[i]] += h[i]
__global__ void pool_kernel(const float* __restrict__ h,
                            const int* __restrict__ batch,
                            float* __restrict__ pooled, int rows) {
    const long long t = (long long)blockIdx.x * blockDim.x + threadIdx.x;
    const int i = (int)(t >> 5);
    const int c = (int)(t & 31);
    if (i < rows) {
        const int g = batch[i];
        const float4 v = *(const float4*)(h + (size_t)i * HID + c * 4);
        float* dp = pooled + (size_t)g * HID + c * 4;
        atomicAdd(dp + 0, v.x);
        atomicAdd(dp + 1, v.y);
        atomicAdd(dp + 2, v.z);
        atomicAdd(dp + 3, v.w);
    }
}

// ---------------------------------------------------------------------------
extern "C" void kernel_launch(void* const* d_in, const int* in_sizes, int n_in,
                              void* d_out, int out_size, void* d_ws, size_t ws_size,
                              hipStream_t stream) {
    const float* x     = (const float*)d_in[0];   // [N,1]
    const int*   ei    = (const int*)d_in[1];     // [2,E]
    const int*   batch = (const int*)d_in[2];     // [N]
    const float* W1_0 = (const float*)d_in[3];  const float* b1_0 = (const float*)d_in[4];
    const float* W2_0 = (const float*)d_in[5];  const float* b2_0 = (const float*)d_in[6];
    const float* W1_1 = (const float*)d_in[7];  const float* b1_1 = (const float*)d_in[8];
    const float* W2_1 = (const float*)d_in[9];  const float* b2_1 = (const float*)d_in[10];
    const float* W1_2 = (const float*)d_in[11]; const float* b1_2 = (const float*)d_in[12];
    const float* W2_2 = (const float*)d_in[13]; const float* b2_2 = (const float*)d_in[14];
    const float* Wfc  = (const float*)d_in[15]; const float* bfc  = (const float*)d_in[16];
    float* out = (float*)d_out;

    const int N = in_sizes[0];         // 100000 (IN == 1)
    const int E = in_sizes[1] / 2;     // 640000
    const int G = out_size / HID;      // 256

    const size_t NH = (size_t)N * HID;
    float* hA   = (float*)d_ws;        // [N,H]
    float* hB   = hA + NH;             // [N,H]
    float* agg  = hB + NH;             // [N,H]; also reused for agg0[N] and pooled[G,H]

    const int mlpBlocks  = (N + 15) / 16;
    const int zeroBlocksNH = (int)((NH + 255) / 256);
    const int scatBlocks = (int)(((long long)E * 32 + 255) / 256);
    const int poolBlocks = (int)(((long long)N * 32 + 255) / 256);

    // ---- Layer 0 (din = 1) ----
    zero_kernel<<<(N + 255) / 256, 256, 0, stream>>>(agg, N);
    scatter0_kernel<<<(E + 255) / 256, 256, 0, stream>>>(x, ei, agg, E);
    layer0_kernel<<<mlpBlocks, 256, 0, stream>>>(x, agg, W1_0, b1_0, W2_0, b2_0, hA, N);

    // ---- Layer 1 ----
    zero_kernel<<<zeroBlocksNH, 256, 0, stream>>>(agg, (long long)NH);
    scatterH_kernel<<<scatBlocks, 256, 0, stream>>>(hA, ei, agg, E);
    mlp_kernel<<<mlpBlocks, 256, 0, stream>>>(hA, agg, W1_1, b1_1, W2_1, b2_1, hB, N);

    // ---- Layer 2 ----
    zero_kernel<<<zeroBlocksNH, 256, 0, stream>>>(agg, (long long)NH);
    scatterH_kernel<<<scatBlocks, 256, 0, stream>>>(hB, ei, agg, E);
    mlp_kernel<<<mlpBlocks, 256, 0, stream>>>(hB, agg, W1_2, b1_2, W2_2, b2_2, hA, N);

    // ---- Global add pool ----
    float* pooled = agg;               // reuse scratch: [G,H]
    zero_kernel<<<((long long)G * HID + 255) / 256, 256, 0, stream>>>(pooled, (long long)G * HID);
    pool_kernel<<<poolBlocks, 256, 0, stream>>>(hA, batch, pooled, N);

    // ---- Final FC ----
    linear_kernel<<<(G + 15) / 16, 256, 0, stream>>>(pooled, Wfc, bfc, out, G);
}